// GNNEncoder_52664888984239
// MI455X (gfx1250) — compile-verified
//
#include <hip/hip_runtime.h>
#include <hip/hip_bf16.h>

#define N_NODES 10000
#define E_EDGES 320000
#define DD      256
#define LL      3
#define LDA     260     // LDS row stride (floats): dword bank = (4m + k) % 64 -> conflict-free

typedef __attribute__((ext_vector_type(2))) float v2f;
typedef __attribute__((ext_vector_type(8))) float v8f;

#define F_RELU 1
#define F_BIAS 2
#define F_RES  4

#define WMMA_F32(a, b, c) \
    __builtin_amdgcn_wmma_f32_16x16x4_f32(false, (a), false, (b), (short)0, (c), false, false)

// ---------------------------------------------------------------------------
// 256x256 transpose (one-time per weight matrix): makes B fragments a single
// 8B load in the WMMA loops.
// ---------------------------------------------------------------------------
__global__ __launch_bounds__(256) void transpose256(const float* __restrict__ in,
                                                    float* __restrict__ out)
{
    __shared__ float tile[32][33];
    int bx = blockIdx.x << 5, by = blockIdx.y << 5;
    int tx = threadIdx.x, ty = threadIdx.y;          // (32, 8)
    #pragma unroll
    for (int i = 0; i < 32; i += 8)
        tile[ty + i][tx] = in[(size_t)(by + ty + i) * DD + bx + tx];
    __syncthreads();
    #pragma unroll
    for (int i = 0; i < 32; i += 8)
        out[(size_t)(bx + ty + i) * DD + by + tx] = tile[tx][ty + i];
}

// Cooperative stage of a 16x256 A slab into padded LDS (float4, coalesced).
__device__ __forceinline__ void stage_A(float* As, const float* __restrict__ A, int row0)
{
    const float4* Ag = (const float4*)(A + (size_t)row0 * DD);
    #pragma unroll
    for (int i = 0; i < 4; ++i) {
        int fidx = threadIdx.x + (i << 8);           // 0..1023 float4s
        int r = fidx >> 6, c4 = fidx & 63;
        float4 t = Ag[fidx];
        *(float4*)&As[r * LDA + (c4 << 2)] = t;
    }
    __syncthreads();
}

// ---------------------------------------------------------------------------
// GEMM: out[16 x 256 per block] = post(A @ W), W given TRANSPOSED (WT[N][K]).
// One wave -> two 16x16 f32 WMMA tiles; A fragments from LDS.
// ---------------------------------------------------------------------------
__global__ __launch_bounds__(256) void gemm256_wmma(
    const float* __restrict__ A, const float* __restrict__ WT,
    const float* __restrict__ bias, const float* __restrict__ resid,
    float* __restrict__ out, int flags)
{
    __shared__ float As[16 * LDA];
    const int row0 = blockIdx.x << 4;
    stage_A(As, A, row0);

    const int wave = threadIdx.x >> 5;
    const int lane = threadIdx.x & 31;
    const int col0 = wave << 5;
    const int m    = lane & 15;
    const int khi  = (lane >> 4) << 1;               // 0 or 2

    const float* Am = As + m * LDA + khi;
    const float* W0 = WT + (size_t)(col0 + m) * DD + khi;
    const float* W1 = W0 + (size_t)16 * DD;
    v8f c0 = {}; v8f c1 = {};

    for (int k = 0; k < DD; k += 4) {
        v2f a  = *(const v2f*)(Am + k);              // ds_load_b64, bank-clean
        v2f b0 = *(const v2f*)(W0 + k);              // global b64
        v2f b1 = *(const v2f*)(W1 + k);
        c0 = WMMA_F32(a, b0, c0);
        c1 = WMMA_F32(a, b1, c1);
    }

    const int mlo = (lane >> 4) << 3;
    const int nn  = lane & 15;
    #pragma unroll
    for (int vi = 0; vi < 8; ++vi) {
        size_t base = (size_t)(row0 + mlo + vi) * DD;
        int cc0 = col0 + nn, cc1 = col0 + 16 + nn;
        float v0 = c0[vi], v1 = c1[vi];
        if (flags & F_BIAS) { v0 += bias[cc0];         v1 += bias[cc1]; }
        if (flags & F_RES)  { v0 += resid[base + cc0]; v1 += resid[base + cc1]; }
        if (flags & F_RELU) { v0 = fmaxf(v0, 0.f);     v1 = fmaxf(v1, 0.f); }
        out[base + cc0] = v0;
        out[base + cc1] = v1;
    }
}

// ---------------------------------------------------------------------------
// Fused QKV: one A stage, 6 WMMA accumulators per wave (3 outputs x 2 tiles).
// ---------------------------------------------------------------------------
__global__ __launch_bounds__(256) void qkv_wmma(
    const float* __restrict__ A,
    const float* __restrict__ WqT, const float* __restrict__ WkT,
    const float* __restrict__ WvT,
    float* __restrict__ q, float* __restrict__ k, float* __restrict__ v)
{
    __shared__ float As[16 * LDA];
    const int row0 = blockIdx.x << 4;
    stage_A(As, A, row0);

    const int wave = threadIdx.x >> 5;
    const int lane = threadIdx.x & 31;
    const int col0 = wave << 5;
    const int m    = lane & 15;
    const int khi  = (lane >> 4) << 1;

    const float* Am = As + m * LDA + khi;
    size_t woff = (size_t)(col0 + m) * DD + khi;
    const float* Q0 = WqT + woff; const float* Q1 = Q0 + (size_t)16 * DD;
    const float* K0 = WkT + woff; const float* K1 = K0 + (size_t)16 * DD;
    const float* V0 = WvT + woff; const float* V1 = V0 + (size_t)16 * DD;

    v8f cq0 = {}, cq1 = {}, ck0 = {}, ck1 = {}, cv0 = {}, cv1 = {};
    for (int kk = 0; kk < DD; kk += 4) {
        v2f a = *(const v2f*)(Am + kk);
        cq0 = WMMA_F32(a, *(const v2f*)(Q0 + kk), cq0);
        cq1 = WMMA_F32(a, *(const v2f*)(Q1 + kk), cq1);
        ck0 = WMMA_F32(a, *(const v2f*)(K0 + kk), ck0);
        ck1 = WMMA_F32(a, *(const v2f*)(K1 + kk), ck1);
        cv0 = WMMA_F32(a, *(const v2f*)(V0 + kk), cv0);
        cv1 = WMMA_F32(a, *(const v2f*)(V1 + kk), cv1);
    }

    const int mlo = (lane >> 4) << 3;
    const int nn  = lane & 15;
    #pragma unroll
    for (int vi = 0; vi < 8; ++vi) {
        size_t base = (size_t)(row0 + mlo + vi) * DD;
        int cc0 = col0 + nn, cc1 = col0 + 16 + nn;
        q[base + cc0] = cq0[vi]; q[base + cc1] = cq1[vi];
        k[base + cc0] = ck0[vi]; k[base + cc1] = ck1[vi];
        v[base + cc0] = cv0[vi]; v[base + cc1] = cv1[vi];
    }
}

// ---------------------------------------------------------------------------
// One wave per edge: scores[e] = <q[dst], k[src]> / sqrt(D)  (b128 gathers)
// ---------------------------------------------------------------------------
__global__ __launch_bounds__(256) void edge_scores(
    const float* __restrict__ q, const float* __restrict__ k,
    const int* __restrict__ edges, float* __restrict__ scores)
{
    int e    = (blockIdx.x << 3) + (threadIdx.x >> 5);
    int lane = threadIdx.x & 31;
    if (e >= E_EDGES) return;
    int src = edges[e];
    int dst = edges[E_EDGES + e];
    const float4* qd = (const float4*)(q + (size_t)dst * DD);
    const float4* ks = (const float4*)(k + (size_t)src * DD);
    float acc = 0.f;
    #pragma unroll
    for (int i = 0; i < 2; ++i) {
        float4 a = qd[lane + (i << 5)];
        float4 b = ks[lane + (i << 5)];
        acc = fmaf(a.x, b.x, acc); acc = fmaf(a.y, b.y, acc);
        acc = fmaf(a.z, b.z, acc); acc = fmaf(a.w, b.w, acc);
    }
    #pragma unroll
    for (int off = 16; off > 0; off >>= 1)
        acc += __shfl_xor(acc, off, 32);
    if (lane == 0) scores[e] = acc * 0.0625f;        // 1/sqrt(256)
}

// monotonic float <-> uint encoding for integer atomicMax on floats
__device__ __forceinline__ unsigned f2ord(float f) {
    unsigned u = __float_as_uint(f);
    return (u & 0x80000000u) ? ~u : (u | 0x80000000u);
}
__device__ __forceinline__ float ord2f(unsigned u) {
    u = (u & 0x80000000u) ? (u & 0x7FFFFFFFu) : ~u;
    return __uint_as_float(u);
}

__global__ void seg_init(unsigned* __restrict__ segmax, float* __restrict__ ssum)
{
    int i = blockIdx.x * blockDim.x + threadIdx.x;
    if (i < N_NODES) { segmax[i] = 0x007FFFFFu; /* ord(-inf) */ ssum[i] = 0.f; }
}

__global__ void seg_max(const float* __restrict__ scores,
                        const int* __restrict__ edges,
                        unsigned* __restrict__ segmax)
{
    int e = blockIdx.x * blockDim.x + threadIdx.x;
    if (e >= E_EDGES) return;
    int dst = edges[E_EDGES + e];
    atomicMax(&segmax[dst], f2ord(scores[e]));
}

__global__ void seg_exp(const float* __restrict__ scores,
                        const int* __restrict__ edges,
                        const unsigned* __restrict__ segmax,
                        float* __restrict__ ssum, float* __restrict__ expv)
{
    int e = blockIdx.x * blockDim.x + threadIdx.x;
    if (e >= E_EDGES) return;
    int dst = edges[E_EDGES + e];
    float t = scores[e] - ord2f(segmax[dst]);
    t = fminf(fmaxf(t, -50.f), 50.f);
    float ex = expf(t);
    expv[e] = ex;
    atomicAdd(&ssum[dst], ex);
}

// ---------------------------------------------------------------------------
// One wave per edge: agg[dst] += v[src] * alpha  (b128 reads, f32 L2 atomics)
// ---------------------------------------------------------------------------
__global__ __launch_bounds__(256) void edge_scatter(
    const float* __restrict__ v, const int* __restrict__ edges,
    const float* __restrict__ expv, const float* __restrict__ ssum,
    float* __restrict__ agg)
{
    int e    = (blockIdx.x << 3) + (threadIdx.x >> 5);
    int lane = threadIdx.x & 31;
    if (e >= E_EDGES) return;
    int src = edges[e];
    int dst = edges[E_EDGES + e];
    float alpha = expv[e] / (ssum[dst] + 1e-9f);
    const float4* vs = (const float4*)(v + (size_t)src * DD);
    float*        ad = agg + (size_t)dst * DD;
    #pragma unroll
    for (int i = 0; i < 2; ++i) {
        int f4 = lane + (i << 5);
        float4 t = vs[f4];
        int c = f4 << 2;
        atomicAdd(&ad[c + 0], t.x * alpha);
        atomicAdd(&ad[c + 1], t.y * alpha);
        atomicAdd(&ad[c + 2], t.z * alpha);
        atomicAdd(&ad[c + 3], t.w * alpha);
    }
}

extern "C" void kernel_launch(void* const* d_in, const int* in_sizes, int n_in,
                              void* d_out, int out_size, void* d_ws, size_t ws_size,
                              hipStream_t stream)
{
    const float* x     = (const float*)d_in[0];
    const int*   edges = (const int*)  d_in[1];
    const float* w_in  = (const float*)d_in[2];
    const float* b_in  = (const float*)d_in[3];
    const float* wq    = (const float*)d_in[4];
    const float* wk    = (const float*)d_in[5];
    const float* wv    = (const float*)d_in[6];
    const float* wp    = (const float*)d_in[7];
    const float* bp    = (const float*)d_in[8];
    float* out = (float*)d_out;

    const size_t nd = (size_t)N_NODES * DD * sizeof(float);
    const size_t wm = (size_t)DD * DD;                    // elements per weight
    char* p = (char*)d_ws;
    float*    h    = (float*)p;    p += nd;
    float*    q    = (float*)p;    p += nd;               // reused as agg
    float*    kk   = (float*)p;    p += nd;
    float*    v    = (float*)p;    p += nd;
    float*    wT   = (float*)p;    p += 13 * wm * sizeof(float);  // transposed weights
    float*    sc   = (float*)p;    p += (size_t)E_EDGES * sizeof(float);
    float*    ex   = (float*)p;    p += (size_t)E_EDGES * sizeof(float);
    unsigned* smax = (unsigned*)p; p += (size_t)N_NODES * sizeof(unsigned);
    float*    ssum = (float*)p;    p += (size_t)N_NODES * sizeof(float);

    float* w_inT = wT;                 // [0]
    float* wqT   = wT + 1 * wm;        // [1..3]
    float* wkT   = wT + 4 * wm;        // [4..6]
    float* wvT   = wT + 7 * wm;        // [7..9]
    float* wpT   = wT + 10 * wm;       // [10..12]

    dim3 tT(32, 8), gT(8, 8);
    dim3 tB(256);
    dim3 gB(N_NODES / 16);             // 625 blocks, exact
    dim3 gE8(E_EDGES / 8);
    dim3 gE((E_EDGES + 255) / 256);
    dim3 gN((N_NODES + 255) / 256);

    // One-time weight transposes (B fragments become single b64 loads).
    transpose256<<<gT, tT, 0, stream>>>(w_in, w_inT);
    for (int l = 0; l < LL; ++l) {
        transpose256<<<gT, tT, 0, stream>>>(wq + (size_t)l * wm, wqT + (size_t)l * wm);
        transpose256<<<gT, tT, 0, stream>>>(wk + (size_t)l * wm, wkT + (size_t)l * wm);
        transpose256<<<gT, tT, 0, stream>>>(wv + (size_t)l * wm, wvT + (size_t)l * wm);
        transpose256<<<gT, tT, 0, stream>>>(wp + (size_t)l * wm, wpT + (size_t)l * wm);
    }

    // h = relu(x @ w_in + b_in)
    gemm256_wmma<<<gB, tB, 0, stream>>>(x, w_inT, b_in, nullptr, h, F_RELU | F_BIAS);

    for (int l = 0; l < LL; ++l) {
        qkv_wmma<<<gB, tB, 0, stream>>>(h,
                                        wqT + (size_t)l * wm,
                                        wkT + (size_t)l * wm,
                                        wvT + (size_t)l * wm,
                                        q, kk, v);

        edge_scores<<<gE8, tB, 0, stream>>>(q, kk, edges, sc);
        seg_init   <<<gN,  tB, 0, stream>>>(smax, ssum);
        seg_max    <<<gE,  tB, 0, stream>>>(sc, edges, smax);
        seg_exp    <<<gE,  tB, 0, stream>>>(sc, edges, smax, ssum, ex);

        hipMemsetAsync(q, 0, nd, stream);             // q becomes agg
        edge_scatter<<<gE8, tB, 0, stream>>>(v, edges, ex, ssum, q);

        float* ho = (l == LL - 1) ? out : h;
        gemm256_wmma<<<gB, tB, 0, stream>>>(q, wpT + (size_t)l * wm,
                                            bp + (size_t)l * DD, h, ho,
                                            F_RELU | F_BIAS | F_RES);
    }
}